// FrameSelection_35897336659982
// MI455X (gfx1250) — compile-verified
//
#include <hip/hip_runtime.h>
#include <hip/hip_bf16.h>
#include <stdint.h>

// Problem constants (fixed by the reference): B=8, T=64, H=W=512
#define NB 8
#define NT 64
#define HW (512 * 512)
#define THREADS_A 256
#define WAVES (THREADS_A / 32)
#define STAGES ((HW / 4) / THREADS_A)   // 256 stages of 256 float4 per block
#define EMPTY_SCORE 1e6f

#if defined(__AMDGCN__) && defined(__gfx1250__) && \
    __has_builtin(__builtin_amdgcn_global_load_async_to_lds_b128) && \
    __has_builtin(__builtin_amdgcn_s_wait_asynccnt)
#define USE_ASYNC 1
#else
#define USE_ASYNC 0
#endif

#define AS1 __attribute__((address_space(1)))
#define AS3 __attribute__((address_space(3)))
typedef int vint4 __attribute__((vector_size(16)));

// ent(x) = -(p ln p + q ln q), p = sigmoid(x). Using ln q = ln p - x:
// ent = softplus(-x) + x * sigmoid(-x); symmetric in x, so evaluate at a=|x|
// where e = exp(-a) <= 1 keeps everything well-conditioned (no inf-inf).
__device__ __forceinline__ void accum_elem(float x, float& cnt, float& tot) {
    float a = fabsf(x);
    float e = __expf(-a);
    float u = 1.0f + e;
    float ent = __logf(u) + __fdividef(a * e, u);
    if (x > 0.0f) {            // strict p > 0.5  <=>  x > 0
        cnt += 1.0f;
        tot += ent;
    }
}

__device__ __forceinline__ void accum_vec4(float4 v, float& cnt, float& tot) {
    accum_elem(v.x, cnt, tot);
    accum_elem(v.y, cnt, tot);
    accum_elem(v.z, cnt, tot);
    accum_elem(v.w, cnt, tot);
}

__global__ __launch_bounds__(THREADS_A)
void FrameScore_kernel(const float* __restrict__ preds, float* __restrict__ score) {
    const int frame = blockIdx.x;                       // b*NT + t, 512 frames
    const float4* fbase = (const float4*)preds + (size_t)frame * (HW / 4);
    const int lane = threadIdx.x & 31;
    const int wave = threadIdx.x >> 5;

    float cnt = 0.0f, tot = 0.0f;

#if USE_ASYNC
    // Per-wave private 8-slot ring of 32x16B staging tiles (32 KB total).
    // Issue depth 4; slot rewrite distance 8 avoids LDS RAW/WAR hazards.
    __shared__ float4 stage[WAVES][8][32];

    #define ISSUE(s)                                                            \
        __builtin_amdgcn_global_load_async_to_lds_b128(                          \
            (AS1 vint4*)(fbase + ((s) * THREADS_A + wave * 32 + lane)),          \
            (AS3 vint4*)(&stage[wave][(s) & 7][lane]), 0, 0)
    #define CONSUME(s)                                                           \
        do { float4 v_ = stage[wave][(s) & 7][lane]; accum_vec4(v_, cnt, tot); } while (0)

    ISSUE(0); ISSUE(1); ISSUE(2); ISSUE(3);
    #pragma unroll 4
    for (int s = 0; s < STAGES - 4; ++s) {
        __builtin_amdgcn_s_wait_asynccnt(3);   // oldest (stage s) has landed in LDS
        CONSUME(s);
        ISSUE(s + 4);
    }
    __builtin_amdgcn_s_wait_asynccnt(2); CONSUME(STAGES - 4);
    __builtin_amdgcn_s_wait_asynccnt(1); CONSUME(STAGES - 3);
    __builtin_amdgcn_s_wait_asynccnt(0); CONSUME(STAGES - 2);
    CONSUME(STAGES - 1);

    #undef ISSUE
    #undef CONSUME
#else
    // Fallback: straight coalesced float4 streaming.
    #pragma unroll 4
    for (int s = 0; s < STAGES; ++s) {
        float4 v = fbase[s * THREADS_A + threadIdx.x];
        accum_vec4(v, cnt, tot);
    }
#endif

    // wave32 butterfly reduction
    #pragma unroll
    for (int off = 16; off > 0; off >>= 1) {
        cnt += __shfl_xor(cnt, off, 32);
        tot += __shfl_xor(tot, off, 32);
    }

    __shared__ float wc[WAVES], wt[WAVES];
    if (lane == 0) { wc[wave] = cnt; wt[wave] = tot; }
    __syncthreads();
    if (threadIdx.x == 0) {
        float C = 0.0f, S = 0.0f;
        #pragma unroll
        for (int w = 0; w < WAVES; ++w) { C += wc[w]; S += wt[w]; }
        score[frame] = (C > 0.0f) ? (S / C) : EMPTY_SCORE;
    }
}

__global__ __launch_bounds__(NT)
void FrameArgmin_kernel(const float* __restrict__ score, int* __restrict__ out) {
    __shared__ float ss[NT];
    __shared__ int   si[NT];
    const int b = blockIdx.x;
    const int t = threadIdx.x;
    ss[t] = score[b * NT + t];
    si[t] = t;
    __syncthreads();
    #pragma unroll
    for (int off = NT / 2; off > 0; off >>= 1) {
        if (t < off) {
            float o = ss[t + off]; int oi = si[t + off];
            // first-occurrence tie-break (matches jnp.argmin)
            if (o < ss[t] || (o == ss[t] && oi < si[t])) { ss[t] = o; si[t] = oi; }
        }
        __syncthreads();
    }
    if (t == 0) out[b] = si[0];
}

extern "C" void kernel_launch(void* const* d_in, const int* in_sizes, int n_in,
                              void* d_out, int out_size, void* d_ws, size_t ws_size,
                              hipStream_t stream) {
    (void)in_sizes; (void)n_in; (void)out_size; (void)ws_size;
    const float* preds = (const float*)d_in[0];     // [8,64,512,512] f32 logits
    float* score = (float*)d_ws;                    // [512] f32 scratch
    int*   out   = (int*)d_out;                     // [8] int32 argmin indices

    FrameScore_kernel<<<NB * NT, THREADS_A, 0, stream>>>(preds, score);
    FrameArgmin_kernel<<<NB, NT, 0, stream>>>(score, out);
}